// GraphEmb_45973329936656
// MI455X (gfx1250) — compile-verified
//
#include <hip/hip_runtime.h>
#include <hip/hip_bf16.h>
#include <math.h>

#define DIM 1024

typedef __attribute__((ext_vector_type(2))) float v2f;
typedef __attribute__((ext_vector_type(8))) float v8f;

// ---------------------------------------------------------------------------
// Kernel 1: per-graph mean of node embeddings.  One block per graph,
// 256 threads, each thread owns 4 consecutive columns (float4 coalesced).
// ---------------------------------------------------------------------------
__global__ void mean_kernel(const float* __restrict__ nodes,  // [B*npg, DIM]
                            float* __restrict__ meanv,        // [B, DIM]
                            int npg)
{
    const int b   = blockIdx.x;
    const int col = threadIdx.x * 4;
    const float* base = nodes + (size_t)b * npg * DIM + col;

    float4 acc = make_float4(0.f, 0.f, 0.f, 0.f);
    for (int n = 0; n < npg; ++n) {
        float4 v = *reinterpret_cast<const float4*>(base + (size_t)n * DIM);
        acc.x += v.x; acc.y += v.y; acc.z += v.z; acc.w += v.w;
    }
    const float inv = 1.0f / (float)npg;
    float4 r = make_float4(acc.x * inv, acc.y * inv, acc.z * inv, acc.w * inv);
    *reinterpret_cast<float4*>(meanv + (size_t)b * DIM + col) = r;
}

// ---------------------------------------------------------------------------
// Kernel 2: common = relu(mean @ theta^T) via V_WMMA_F32_16X16X4_F32.
// C[M=b][N=d] = sum_e mean[b][e] * theta[d][e]  (B matrix == theta row-major,
// indexed [n][k], i.e. already the transpose we need).
//
// A 16x4 f32 layout (ISA 7.12.2): lanes 0-15 -> M=lane, K=0(v0)/1(v1);
// lanes 16-31 -> M=lane-16, K=2(v0)/3(v1).  B mirrors with N in place of M.
// C/D 16x16 f32: lanes 0-15 -> N=lane, M=vgpr; lanes 16-31 -> N=lane-16, M=vgpr+8.
// One wave computes one 16x16 tile; uniform per-wave control flow keeps EXEC
// all-ones around the WMMA as required.
// ---------------------------------------------------------------------------
__global__ void gemm_relu_wmma(const float* __restrict__ meanv,  // [B, DIM]
                               const float* __restrict__ theta,  // [DIM, DIM]
                               float* __restrict__ common,       // [B, DIM]
                               int num_tiles)
{
    const int lane = threadIdx.x & 31;
    const int wave = threadIdx.x >> 5;
    const int tile = blockIdx.x * (blockDim.x >> 5) + wave;
    if (tile >= num_tiles) return;   // whole-wave uniform exit

    const int tilesN = DIM / 16;
    const int tm = (tile / tilesN) * 16;   // graph-row base
    const int tn = (tile % tilesN) * 16;   // output-dim base

    const int l15  = lane & 15;
    const int half = lane >> 4;            // 0 or 1
    const int kb   = half << 1;            // K sub-offset 0 or 2

    const float* arow = meanv + (size_t)(tm + l15) * DIM + kb;
    const float* brow = theta + (size_t)(tn + l15) * DIM + kb;

    v8f c = {};
    for (int k = 0; k < DIM; k += 4) {
        v2f a, b;
        a.x = arow[k];
        a.y = arow[k + 1];
        b.x = brow[k];
        b.y = brow[k + 1];
        c = __builtin_amdgcn_wmma_f32_16x16x4_f32(
                /*neg_a=*/false, a, /*neg_b=*/false, b,
                /*c_mod=*/(short)0, c, /*reuse_a=*/false, /*reuse_b=*/false);
    }

#pragma unroll
    for (int i = 0; i < 8; ++i) {
        const int m = half * 8 + i;
        const float v = c[i];
        common[(size_t)(tm + m) * DIM + tn + l15] = fmaxf(v, 0.0f);
    }
}

// ---------------------------------------------------------------------------
// Kernel 3: per-node sigmoid gate + gated per-graph mean.
// One block per graph (256 threads = 8 waves).  common[b] staged in LDS;
// each wave computes dots for nodes wave, wave+8, ... with a wave32 shuffle
// reduction; then all threads do the gated float4 accumulation.
// ---------------------------------------------------------------------------
__global__ void gate_sum_kernel(const float* __restrict__ nodes,   // [B*npg, DIM]
                                const float* __restrict__ common,  // [B, DIM]
                                float* __restrict__ out,           // [B, DIM]
                                int npg)
{
    __shared__ float s_common[DIM];
    __shared__ float s_gate[256];

    const int b    = blockIdx.x;
    const int t    = threadIdx.x;
    const int lane = t & 31;
    const int wave = t >> 5;

    for (int i = t; i < DIM; i += 256)
        s_common[i] = common[(size_t)b * DIM + i];
    __syncthreads();

    const float* gbase = nodes + (size_t)b * npg * DIM;

    // Phase 1: gates (8 waves x strided nodes)
    for (int n = wave; n < npg; n += 8) {
        const float* row = gbase + (size_t)n * DIM;
        float p = 0.f;
        for (int i = lane; i < DIM; i += 32)
            p += row[i] * s_common[i];
        // wave32 tree reduction
        for (int off = 16; off > 0; off >>= 1)
            p += __shfl_down(p, off, 32);
        if (lane == 0)
            s_gate[n] = 1.0f / (1.0f + __expf(-p));
    }
    __syncthreads();

    // Phase 2: gated mean, 4 columns per thread
    const int col = t * 4;
    float4 acc = make_float4(0.f, 0.f, 0.f, 0.f);
    for (int n = 0; n < npg; ++n) {
        if (n + 1 < npg)
            __builtin_prefetch(gbase + (size_t)(n + 1) * DIM + col, 0, 1);
        const float g = s_gate[n];
        float4 v = *reinterpret_cast<const float4*>(gbase + (size_t)n * DIM + col);
        acc.x += g * v.x; acc.y += g * v.y; acc.z += g * v.z; acc.w += g * v.w;
    }
    const float inv = 1.0f / (float)npg;
    float4 r = make_float4(acc.x * inv, acc.y * inv, acc.z * inv, acc.w * inv);
    *reinterpret_cast<float4*>(out + (size_t)b * DIM + col) = r;
}

// ---------------------------------------------------------------------------
// Launch: mean -> wmma GEMM(+relu) -> gate+gated-mean
// Inputs (reference order): eb_nodes[N,D] f32, eb_edges[B,D] f32 (unused),
// numb_nodes[B] (unused, uniform), numb_edges[B] (unused), theta[D,D] f32.
// ---------------------------------------------------------------------------
extern "C" void kernel_launch(void* const* d_in, const int* in_sizes, int n_in,
                              void* d_out, int out_size, void* d_ws, size_t ws_size,
                              hipStream_t stream)
{
    const float* nodes = (const float*)d_in[0];
    const float* theta = (const float*)d_in[4];

    const int Bg   = in_sizes[1] / DIM;            // 512 (from eb_edges [B,D])
    const int Ntot = in_sizes[0] / DIM;            // 32768
    const int npg  = Ntot / Bg;                    // 64

    float* meanv  = (float*)d_ws;                  // [Bg, DIM]
    float* common = meanv + (size_t)Bg * DIM;      // [Bg, DIM]

    mean_kernel<<<Bg, 256, 0, stream>>>(nodes, meanv, npg);

    const int num_tiles = (Bg / 16) * (DIM / 16);  // 2048 16x16 C tiles
    const int wavesPerBlock = 8;                   // 256 threads
    const int gemmBlocks = (num_tiles + wavesPerBlock - 1) / wavesPerBlock;
    gemm_relu_wmma<<<gemmBlocks, 256, 0, stream>>>(meanv, theta, common, num_tiles);

    gate_sum_kernel<<<Bg, 256, 0, stream>>>(nodes, common, (float*)d_out, npg);
}